// RunningCenters_30829275250754
// MI455X (gfx1250) — compile-verified
//
#include <hip/hip_runtime.h>

// ---------------------------------------------------------------------------
// RunningCenters on MI455X (gfx1250)
//   sums[c][e]  += x[i][e]  for y[i]==c        (f32 atomics, L2-resident 512KB)
//   counts[c]   += 1
//   out = present ? (sums/count + centers*nbt)/(nbt+1) : centers
// x (256 MB, read once) is streamed via async global->LDS B128 copies with
// th:TH_LOAD_NT (don't thrash L2 where the atomic accumulator lives),
// double-buffered per wave, tracked with ASYNCcnt (CDNA5 async path).
// ---------------------------------------------------------------------------

#define N_EMBED   128
#define N_CLASSES 1000
#define ROWS      4                       // rows per async batch per wave
#define ROW_BYTES (N_EMBED * 4)           // 512 B  (32 lanes x 16 B)
#define BUF_BYTES (ROWS * ROW_BYTES)      // 2 KB per buffer
#define WAVES_PER_BLOCK 8
#define BLOCK     256
#define GRID_SCATTER 2048                 // 16384 waves -> 8 sweeps over N

// LDS byte offset of a generic pointer known to point into LDS.
__device__ __forceinline__ unsigned lds_addr_of(const void* p) {
  return (unsigned)(unsigned long long)
      (__attribute__((address_space(3))) const char*)p;
}

// One wave moves one 512B row: 32 lanes x 16 bytes, async into LDS.
// Non-temporal: x is a one-shot stream; keep it out of L2 so the 512 KB
// atomic accumulator stays resident.
__device__ __forceinline__ void async_row_load(const float* g, unsigned lds_byte) {
  asm volatile("global_load_async_to_lds_b128 %0, %1, off th:TH_LOAD_NT"
               :: "v"(lds_byte), "v"((unsigned long long)g)
               : "memory");
}

__device__ __forceinline__ void issue_batch(const float* __restrict__ x,
                                            unsigned s, unsigned n,
                                            unsigned lds_row0, int lane) {
#pragma unroll
  for (int r = 0; r < ROWS; ++r) {
    if (s + (unsigned)r < n) {
      const float* g = x + (size_t)(s + r) * N_EMBED + lane * 4;
      async_row_load(g, lds_row0 + r * ROW_BYTES + lane * 16);
    }
  }
}

__global__ __launch_bounds__(BLOCK) void scatter_kernel(
    const float* __restrict__ x, const int* __restrict__ y,
    float* __restrict__ sums, float* __restrict__ counts, unsigned n)
{
  __shared__ __align__(16) char smem[WAVES_PER_BLOCK * 2 * BUF_BYTES]; // 32 KB

  const int      lane   = threadIdx.x & 31;
  const int      wv     = threadIdx.x >> 5;
  const unsigned gw     = blockIdx.x * WAVES_PER_BLOCK + wv;
  const unsigned nWaves = gridDim.x * WAVES_PER_BLOCK;
  const unsigned sweep  = nWaves * ROWS;

  char*          mybuf = smem + wv * 2 * BUF_BYTES;
  const unsigned lds0  = lds_addr_of(mybuf);

  unsigned s = gw * ROWS;
  if (s >= n) return;

  issue_batch(x, s, n, lds0, lane);                       // prime buffer 0
  int cur = 0;
  for (;;) {
    const unsigned snext = s + sweep;
    const bool     more  = (snext < n);
    if (more) {
      issue_batch(x, snext, n, lds0 + (cur ^ 1) * BUF_BYTES, lane);
      // 8 asyncs outstanding; loads complete in order, so <=4 left
      // guarantees the current buffer's 4 rows have landed in LDS.
      asm volatile("s_wait_asynccnt 4" ::: "memory");
    } else {
      asm volatile("s_wait_asynccnt 0" ::: "memory");
    }

    const float* buf = (const float*)(mybuf + cur * BUF_BYTES);
#pragma unroll
    for (int r = 0; r < ROWS; ++r) {
      const unsigned samp = s + r;
      if (samp < n) {
        const int    c = __builtin_nontemporal_load(y + samp);
        const float4 v = *(const float4*)(buf + r * N_EMBED + lane * 4);
        float* dst = sums + (size_t)c * N_EMBED + lane * 4;
        unsafeAtomicAdd(dst + 0, v.x);
        unsafeAtomicAdd(dst + 1, v.y);
        unsafeAtomicAdd(dst + 2, v.z);
        unsafeAtomicAdd(dst + 3, v.w);
        if (lane == 0) unsafeAtomicAdd(counts + c, 1.0f);
      }
    }
    if (!more) break;
    s = snext;
    cur ^= 1;
  }
}

__global__ void zero_kernel(float* __restrict__ sums, float* __restrict__ counts) {
  const int i = blockIdx.x * BLOCK + threadIdx.x;
  if (i < N_CLASSES * N_EMBED) sums[i] = 0.0f;
  if (i < N_CLASSES)           counts[i] = 0.0f;
}

// In-place: d_out currently holds sums; rewrite with updated centers.
__global__ void finalize_kernel(float* __restrict__ io_sums_out,
                                const float* __restrict__ counts,
                                const float* __restrict__ centers,
                                const float* __restrict__ nbt_ptr)
{
  const int i = blockIdx.x * BLOCK + threadIdx.x;
  if (i >= N_CLASSES * N_EMBED) return;
  const int   c   = i >> 7;                 // N_EMBED == 128
  const float cnt = counts[c];
  const float ctr = centers[i];
  const float nbt = nbt_ptr[0];
  float r = ctr;
  if (cnt > 0.0f) {
    const float mu = io_sums_out[i] / cnt;  // cnt>0 so max(cnt,1)==cnt
    r = (mu + ctr * nbt) / (nbt + 1.0f);
  }
  io_sums_out[i] = r;
}

extern "C" void kernel_launch(void* const* d_in, const int* in_sizes, int n_in,
                              void* d_out, int out_size, void* d_ws, size_t ws_size,
                              hipStream_t stream) {
  const float* x       = (const float*)d_in[0];
  const int*   y       = (const int*)  d_in[1];
  const float* centers = (const float*)d_in[2];
  const float* nbt     = (const float*)d_in[3];

  float* sums   = (float*)d_out;   // 128000 f32 accumulator, finalized in place
  float* counts = (float*)d_ws;    // 1000 f32

  const unsigned n = (unsigned)in_sizes[1];
  const int grid_e = (N_CLASSES * N_EMBED + BLOCK - 1) / BLOCK;

  zero_kernel   <<<grid_e,       BLOCK, 0, stream>>>(sums, counts);
  scatter_kernel<<<GRID_SCATTER, BLOCK, 0, stream>>>(x, y, sums, counts, n);
  finalize_kernel<<<grid_e,      BLOCK, 0, stream>>>(sums, counts, centers, nbt);
}